// EdgeConv_SDP_52484500357665
// MI455X (gfx1250) — compile-verified
//
#include <hip/hip_runtime.h>
#include <math.h>

// ---------------------------------------------------------------------------
// CDNA5 / gfx1250 implementation of the EdgeConv GNN forward pass.
// All GEMMs run on V_WMMA_F32_16X16X4_F32 (fp32 matrix pipe, wave32).
// BatchNorm(batch-stats) -> Linear is folded into a single GEMM with
// runtime-adjusted weights. Column sum/sumsq stats for the NEXT BatchNorm
// are fused into the GEMM epilogue (saves a full re-read of every GEMM
// output). segment_sum uses global_atomic_add_f32.
// ---------------------------------------------------------------------------

typedef float v2f __attribute__((ext_vector_type(2)));
typedef float v8f __attribute__((ext_vector_type(8)));

#define BN_EPS 1e-5f
#define KT 16          // K chunk staged through LDS (4 wmma steps of K=4)

// ============================ WMMA GEMM =====================================
// C[M, Npad] = act(A[M, Kpad] @ W[Kpad, Npad] + bias)
// Block: 256 threads = 8 waves arranged 4(M) x 2(N); block tile 64x32.
// Each wave owns one 16x16 C tile, accumulated in an 8-VGPR v8f fragment.
// If doStats: also accumulates per-column sum / sum-of-squares of the stored
// (post-activation) C into sums/sq via LDS + global f32 atomics.
__global__ __launch_bounds__(256) void wmma_gemm_kernel(
    const float* __restrict__ A, int lda, int M,
    const float* __restrict__ W, const float* __restrict__ bias,
    float* __restrict__ C, int ldc, int Kpad, int ldw, int act,
    float* __restrict__ sums, float* __restrict__ sq, int doStats)
{
  __shared__ float As[64][KT + 1];   // +1 pad: distinct banks for column reads
  __shared__ float Bs[KT][32];
  __shared__ float sSum[32];
  __shared__ float sSq[32];

  const int tid  = threadIdx.x;
  const int lane = tid & 31;
  const int wave = tid >> 5;
  const int wm   = wave >> 1;        // 0..3 (M tile within block)
  const int wn   = wave & 1;         // 0..1 (N tile within block)
  const int lrow = lane & 15;
  const int hi   = lane >> 4;        // half-wave selector (K pairs / rows+8)
  const int row0 = blockIdx.x * 64;
  const int col0 = blockIdx.y * 32;

  if (tid < 32) { sSum[tid] = 0.f; sSq[tid] = 0.f; }

  v8f acc = {0.f, 0.f, 0.f, 0.f, 0.f, 0.f, 0.f, 0.f};

  const int ar = tid >> 2;           // 0..63  : A panel row this thread loads
  const int aq = (tid & 3) * 4;      // 0,4,8,12
  const int br = tid >> 3;           // 0..15  : B panel row (tid < 128)
  const int bq = (tid & 7) * 4;

  for (int k0 = 0; k0 < Kpad; k0 += KT) {
    // ---- stage A panel (64 x 16) : one float4 per thread ----
    float4 av = {0.f, 0.f, 0.f, 0.f};
    const int gr = row0 + ar;
    if (gr < M) av = *(const float4*)(A + (size_t)gr * lda + k0 + aq);
    __syncthreads();   // previous iteration's frags fully consumed
    As[ar][aq + 0] = av.x; As[ar][aq + 1] = av.y;
    As[ar][aq + 2] = av.z; As[ar][aq + 3] = av.w;
    // ---- stage B panel (16 x 32) : threads 0..127, one float4 each ----
    if (tid < 128) {
      float4 bv = *(const float4*)(W + (size_t)(k0 + br) * ldw + col0 + bq);
      Bs[br][bq + 0] = bv.x; Bs[br][bq + 1] = bv.y;
      Bs[br][bq + 2] = bv.z; Bs[br][bq + 3] = bv.w;
    }
    __syncthreads();

    // ---- 4 x V_WMMA_F32_16X16X4_F32 ----
    // A frag layout: lanes 0-15 rows 0-15 {K=k,k+1}; lanes 16-31 {K=k+2,k+3}
    // B frag layout: lanes 0-15 cols 0-15 rows {k,k+1}; lanes 16-31 {k+2,k+3}
#pragma unroll
    for (int kk = 0; kk < KT; kk += 4) {
      v2f a, b;
      a.x = As[wm * 16 + lrow][kk + hi * 2 + 0];
      a.y = As[wm * 16 + lrow][kk + hi * 2 + 1];
      b.x = Bs[kk + hi * 2 + 0][wn * 16 + lrow];
      b.y = Bs[kk + hi * 2 + 1][wn * 16 + lrow];
      acc = __builtin_amdgcn_wmma_f32_16x16x4_f32(
          false, a, false, b, (short)0, acc, false, false);
    }
  }
  __syncthreads();

  // Epilogue: C frag VGPR r -> row (r | r+8 for hi half), lane -> column.
  const int cloc  = wn * 16 + lrow;          // column within block tile
  const int col   = col0 + cloc;
  const float bv  = bias[col];
  const int rbase = row0 + wm * 16 + hi * 8;
  float ps = 0.f, psq = 0.f;
#pragma unroll
  for (int r = 0; r < 8; ++r) {
    const int grr = rbase + r;
    if (grr < M) {
      float v = acc[r] + bv;
      if (act == 1) v = fmaxf(v, 0.f);
      C[(size_t)grr * ldc + col] = v;
      ps += v; psq += v * v;
    }
  }
  if (doStats) {
    atomicAdd(&sSum[cloc], ps);
    atomicAdd(&sSq[cloc], psq);
    __syncthreads();
    if (tid < 32) {
      atomicAdd(&sums[col0 + tid], sSum[tid]);
      atomicAdd(&sq[col0 + tid], sSq[tid]);
    }
  }
}

// ======== column mean / sumsq reduction (for gather-built tensors) ==========
__global__ __launch_bounds__(256) void colstats_kernel(
    const float* __restrict__ X, int ld, int M, int ncols,
    float* __restrict__ sums, float* __restrict__ sq)
{
  const int c0 = threadIdx.x;
  const int c1 = c0 + 256;
  const bool h0 = c0 < ncols, h1 = c1 < ncols;
  float s0 = 0.f, q0 = 0.f, s1 = 0.f, q1 = 0.f;
  for (int r = blockIdx.x; r < M; r += gridDim.x) {
    const float* row = X + (size_t)r * ld;
    if (h0) { float v = row[c0]; s0 += v; q0 += v * v; }
    if (h1) { float v = row[c1]; s1 += v; q1 += v * v; }
  }
  if (h0) { atomicAdd(&sums[c0], s0); atomicAdd(&sq[c0], q0); }
  if (h1) { atomicAdd(&sums[c1], s1); atomicAdd(&sq[c1], q1); }
}

// ================= fold BN (batch stats) into following Linear ==============
// Wf[k][j] = s_k * W[k][j]  (zero-padded to Kpad x Npad)
// bf[j]    = b[j] + sum_k t_k * W[k][j],  s=g*rsqrt(var+eps), t=beta-m*s
// bng==nullptr -> identity (plain Linear, just pad).
__global__ void fold_bn_lin_kernel(
    const float* __restrict__ sums, const float* __restrict__ sq, float invM,
    const float* __restrict__ bng, const float* __restrict__ bnb,
    const float* __restrict__ W, const float* __restrict__ lb,
    int K, int Nc, int Kpad, int Npad,
    float* __restrict__ Wf, float* __restrict__ bf)
{
  const int j = blockIdx.x * blockDim.x + threadIdx.x;
  if (j >= Npad) return;
  float acc = 0.f;
  for (int k = 0; k < Kpad; ++k) {
    float wf = 0.f;
    if (k < K && j < Nc) {
      const float w = W[(size_t)k * Nc + j];
      float s = 1.f, t = 0.f;
      if (bng) {
        const float m  = sums[k] * invM;
        const float vv = sq[k] * invM - m * m;
        const float rs = rsqrtf(vv + BN_EPS);
        s = bng[k] * rs;
        t = bnb[k] - m * s;
      }
      wf = s * w;
      acc += t * w;
    }
    Wf[(size_t)k * Npad + j] = wf;
  }
  bf[j] = (j < Nc) ? (lb[j] + acc) : 0.f;
}

// ================= trailing BN + segment_sum scatter (atomic) ===============
__global__ __launch_bounds__(256) void bn_scatter_kernel(
    const float* __restrict__ H, int ld, int E, int dim,
    const float* __restrict__ sums, const float* __restrict__ sq, float invM,
    const float* __restrict__ g, const float* __restrict__ b,
    const int* __restrict__ dstIdx, float* __restrict__ Xout, int ldo)
{
  const int gid = blockIdx.x * blockDim.x + threadIdx.x;
  if (gid >= E * dim) return;
  const int e = gid / dim, j = gid - e * dim;
  const float m  = sums[j] * invM;
  const float vv = sq[j] * invM - m * m;
  const float rs = rsqrtf(vv + BN_EPS);
  const float v  = g[j] * (H[(size_t)e * ld + j] - m) * rs + b[j];
  atomicAdd(&Xout[(size_t)dstIdx[e] * ldo + j], v);
}

// ===================== trailing BN, elementwise (emm1 out) ==================
__global__ __launch_bounds__(256) void bn_apply_kernel(
    const float* __restrict__ H, int ld, int E, int dim,
    const float* __restrict__ sums, const float* __restrict__ sq, float invM,
    const float* __restrict__ g, const float* __restrict__ b,
    float* __restrict__ Out, int ldo)
{
  const int gid = blockIdx.x * blockDim.x + threadIdx.x;
  if (gid >= E * dim) return;
  const int e = gid / dim, j = gid - e * dim;
  const float m  = sums[j] * invM;
  const float vv = sq[j] * invM - m * m;
  const float rs = rsqrtf(vv + BN_EPS);
  Out[(size_t)e * ldo + j] = g[j] * (H[(size_t)e * ld + j] - m) * rs + b[j];
}

// ============================ concat builders ===============================
__global__ __launch_bounds__(256) void build_h0_kernel(
    const float* __restrict__ nf, const int* __restrict__ ei, int E,
    float* __restrict__ H)
{
  const int gid = blockIdx.x * blockDim.x + threadIdx.x;
  if (gid >= E * 32) return;
  const int e = gid >> 5, j = gid & 31;
  const int src = ei[e], dst = ei[E + e];
  float v;
  if (j < 16) v = nf[(size_t)dst * 32 + j];
  else { const int jj = j - 16; v = nf[(size_t)src * 32 + jj] - nf[(size_t)dst * 32 + jj]; }
  H[(size_t)e * 32 + j] = v;
}

__global__ __launch_bounds__(256) void build_g0_kernel(
    const float* __restrict__ ef, const int* __restrict__ ei, int E,
    const float* __restrict__ X1, float* __restrict__ G)
{
  const int gid = blockIdx.x * blockDim.x + threadIdx.x;
  if (gid >= E * 160) return;
  const int e = gid / 160, j = gid - e * 160;
  const int src = ei[e], dst = ei[E + e];
  float v = 0.f;
  if (j < 19)       v = ef[(size_t)e * 38 + j];
  else if (j < 83)  v = X1[(size_t)src * 64 + (j - 19)];
  else if (j < 147) v = X1[(size_t)dst * 64 + (j - 83)];
  G[(size_t)e * 160 + j] = v;
}

__global__ __launch_bounds__(256) void build_c0_kernel(
    const int* __restrict__ ei, int E, const float* __restrict__ X1,
    float* __restrict__ Cb)
{
  const int gid = blockIdx.x * blockDim.x + threadIdx.x;
  if (gid >= E * 128) return;
  const int e = gid >> 7, j = gid & 127;
  const int src = ei[e], dst = ei[E + e];
  float v;
  if (j < 64) v = X1[(size_t)dst * 64 + j];
  else { const int jj = j - 64; v = X1[(size_t)src * 64 + jj] - X1[(size_t)dst * 64 + jj]; }
  Cb[(size_t)e * 128 + j] = v;
}

__global__ __launch_bounds__(256) void build_m0_kernel(
    const float* __restrict__ E1b, const float* __restrict__ X2,
    const int* __restrict__ ei, int E, float* __restrict__ M0)
{
  const int gid = blockIdx.x * blockDim.x + threadIdx.x;
  if (gid >= E * 336) return;
  const int e = gid / 336, j = gid - e * 336;
  const int src = ei[e], dst = ei[E + e];
  float v = 0.f;
  if (j < 76)       v = E1b[(size_t)e * 96 + j];
  else if (j < 204) v = X2[(size_t)src * 128 + (j - 76)];
  else if (j < 332) v = X2[(size_t)dst * 128 + (j - 204)];
  M0[(size_t)e * 336 + j] = v;
}

// =================== final (cols -> 1) dot + sigmoid ========================
__global__ __launch_bounds__(256) void dot_sigmoid_kernel(
    const float* __restrict__ X, int ld, int cols, int M,
    const float* __restrict__ w, const float* __restrict__ b,
    float* __restrict__ out)
{
  __shared__ float red[256];
  const int row = blockIdx.x;
  if (row >= M) return;
  float acc = 0.f;
  for (int c = threadIdx.x; c < cols; c += 256)
    acc += X[(size_t)row * ld + c] * w[c];
  red[threadIdx.x] = acc;
  __syncthreads();
  for (int s = 128; s > 0; s >>= 1) {
    if (threadIdx.x < s) red[threadIdx.x] += red[threadIdx.x + s];
    __syncthreads();
  }
  if (threadIdx.x == 0)
    out[row] = 1.f / (1.f + expf(-(red[0] + b[0])));
}

// ============================== host driver =================================
static inline int ceilDiv(int a, int b) { return (a + b - 1) / b; }

extern "C" void kernel_launch(void* const* d_in, const int* in_sizes, int n_in,
                              void* d_out, int out_size, void* d_ws, size_t ws_size,
                              hipStream_t stream)
{
  if (n_in < 75) return;
  const float* nf = (const float*)d_in[0];   // [N, 32]  (first 16 used)
  const float* ef = (const float*)d_in[1];   // [E, 38]  (first 19 used)
  const int*   ei = (const int*)d_in[2];     // [2, E]: row0=src, row1=dst
  const int E = in_sizes[2] / 2;
  const int N = in_sizes[0] / 32;
  const float invE = 1.f / (float)E;

  auto F = [&](int i) -> const float* { return (const float*)d_in[i]; };
  // pytree flatten order: ehead, emm1, emm2, nhead, nmm1, nmm2 (sorted keys);
  // lin dict {'W','b'} -> W,b ; bn dict {'g','b'} -> b,g (sorted keys).
  const float *ehW1=F(3),*ehB1=F(4),*ehW2=F(5),*ehB2=F(6),*ehW3=F(7),*ehB3=F(8),
              *ehW4=F(9),*ehB4=F(10),*ehW5=F(11),*ehB5=F(12);
  const float *e1bn0b=F(13),*e1bn0g=F(14),*e1W1=F(15),*e1B1=F(16),
              *e1bn1b=F(17),*e1bn1g=F(18),*e1W2=F(19),*e1B2=F(20),
              *e1bn2b=F(21),*e1bn2g=F(22),*e1W3=F(23),*e1B3=F(24),
              *e1bn3b=F(25),*e1bn3g=F(26);
  const float *e2bn0b=F(27),*e2bn0g=F(28),*e2W1=F(29),*e2B1=F(30),
              *e2bn1b=F(31),*e2bn1g=F(32),*e2W2=F(33),*e2B2=F(34),
              *e2bn2b=F(35),*e2bn2g=F(36),*e2W3=F(37),*e2B3=F(38),
              *e2bn3b=F(39),*e2bn3g=F(40);
  const float *nhW1=F(41),*nhB1=F(42),*nhW2=F(43),*nhB2=F(44),
              *nhW3=F(45),*nhB3=F(46),*nhW4=F(47),*nhB4=F(48);
  const float *n1bn0b=F(49),*n1bn0g=F(50),*n1W1=F(51),*n1B1=F(52),
              *n1bn1b=F(53),*n1bn1g=F(54),*n1W2=F(55),*n1B2=F(56),
              *n1bn2b=F(57),*n1bn2g=F(58),*n1W3=F(59),*n1B3=F(60),
              *n1bn3b=F(61),*n1bn3g=F(62);
  const float *n2W1=F(63),*n2B1=F(64),*n2bn1b=F(65),*n2bn1g=F(66),
              *n2W2=F(67),*n2B2=F(68),*n2bn2b=F(69),*n2bn2g=F(70),
              *n2W3=F(71),*n2B3=F(72),*n2bn3b=F(73),*n2bn3g=F(74);

  // -------- workspace layout (floats) --------
  float* ws = (float*)d_ws;
  size_t off = 0;
  float* BUFA = ws + off; off += (size_t)E * 336;   // widest: emm2 input
  float* BUFB = ws + off; off += (size_t)E * 320;   // widest: ehead hidden
  float* E1B  = ws + off; off += (size_t)E * 96;    // emm1 output (76, ld 96)
  float* X1   = ws + off; off += (size_t)N * 64;
  float* X2   = ws + off; off += (size_t)N * 128;
  float* SUM  = ws + off; off += 512;
  float* SQ   = ws + off; off += 512;
  float* BF   = ws + off; off += 512;
  float* WF   = ws + off; off += 131072;            // max folded W: 320x320
  if (ws_size < off * sizeof(float)) return;        // insufficient scratch

  float* outNode = (float*)d_out;
  float* outEdge = (float*)d_out + N;

  // NOTE stream ordering: fold (consumer of SUM/SQ) is launched BEFORE the
  // memset that clears them for the next layer's fused stats.
  auto statreset = [&]() {
    hipMemsetAsync(SUM, 0, 512 * sizeof(float), stream);
    hipMemsetAsync(SQ,  0, 512 * sizeof(float), stream);
  };
  auto stats = [&](const float* X, int ld, int Mr, int ncols) {
    statreset();
    colstats_kernel<<<512, 256, 0, stream>>>(X, ld, Mr, ncols, SUM, SQ);
  };
  auto fold = [&](const float* g, const float* b, const float* W,
                  const float* lb, int K, int Nc, int Kpad, int Npad) {
    fold_bn_lin_kernel<<<ceilDiv(Npad, 128), 128, 0, stream>>>(
        SUM, SQ, invE, g, b, W, lb, K, Nc, Kpad, Npad, WF, BF);
  };
  auto gemm = [&](const float* A, int lda, int Mr, float* C, int Kpad,
                  int Npad, int act, int doStats) {
    if (doStats) statreset();          // runs after the preceding fold
    dim3 grid(ceilDiv(Mr, 64), Npad / 32);
    wmma_gemm_kernel<<<grid, 256, 0, stream>>>(A, lda, Mr, WF, BF, C, Npad,
                                               Kpad, Npad, act, SUM, SQ,
                                               doStats);
  };

  hipMemsetAsync(X1, 0, (size_t)N * 64 * sizeof(float), stream);
  hipMemsetAsync(X2, 0, (size_t)N * 128 * sizeof(float), stream);

  // ======== stage A: nmm1 edge-conv: x1 = segsum(BN(mlp(concat))) =========
  build_h0_kernel<<<ceilDiv(E * 32, 256), 256, 0, stream>>>(nf, ei, E, BUFA);
  stats(BUFA, 32, E, 32);
  fold(n1bn0g, n1bn0b, n1W1, n1B1, 32, 64, 32, 64);
  gemm(BUFA, 32, E, BUFB, 32, 64, 1, 1);
  fold(n1bn1g, n1bn1b, n1W2, n1B2, 64, 64, 64, 64);
  gemm(BUFB, 64, E, BUFA, 64, 64, 1, 1);
  fold(n1bn2g, n1bn2b, n1W3, n1B3, 64, 64, 64, 64);
  gemm(BUFA, 64, E, BUFB, 64, 64, 1, 1);
  bn_scatter_kernel<<<ceilDiv(E * 64, 256), 256, 0, stream>>>(
      BUFB, 64, E, 64, SUM, SQ, invE, n1bn3g, n1bn3b, ei + E, X1, 64);

  // ======== stage B: emm1 edge update -> E1B [E,76 (ld 96)] ===============
  build_g0_kernel<<<ceilDiv(E * 160, 256), 256, 0, stream>>>(ef, ei, E, X1, BUFA);
  stats(BUFA, 160, E, 147);
  fold(e1bn0g, e1bn0b, e1W1, e1B1, 147, 76, 160, 96);
  gemm(BUFA, 160, E, BUFB, 160, 96, 1, 1);
  fold(e1bn1g, e1bn1b, e1W2, e1B2, 76, 76, 96, 96);
  gemm(BUFB, 96, E, BUFA, 96, 96, 1, 1);
  fold(e1bn2g, e1bn2b, e1W3, e1B3, 76, 76, 96, 96);
  gemm(BUFA, 96, E, BUFB, 96, 96, 1, 1);
  bn_apply_kernel<<<ceilDiv(E * 76, 256), 256, 0, stream>>>(
      BUFB, 96, E, 76, SUM, SQ, invE, e1bn3g, e1bn3b, E1B, 96);

  // ======== stage C: nmm2 edge-conv -> X2 [N,128] =========================
  build_c0_kernel<<<ceilDiv(E * 128, 256), 256, 0, stream>>>(ei, E, X1, BUFA);
  fold(nullptr, nullptr, n2W1, n2B1, 128, 128, 128, 128);
  gemm(BUFA, 128, E, BUFB, 128, 128, 1, 1);
  fold(n2bn1g, n2bn1b, n2W2, n2B2, 128, 128, 128, 128);
  gemm(BUFB, 128, E, BUFA, 128, 128, 1, 1);
  fold(n2bn2g, n2bn2b, n2W3, n2B3, 128, 128, 128, 128);
  gemm(BUFA, 128, E, BUFB, 128, 128, 1, 1);
  bn_scatter_kernel<<<ceilDiv(E * 128, 256), 256, 0, stream>>>(
      BUFB, 128, E, 128, SUM, SQ, invE, n2bn3g, n2bn3b, ei + E, X2, 128);

  // ======== stage D: emm2 edge update (trailing BN folds into ehead) ======
  build_m0_kernel<<<ceilDiv(E * 336, 256), 256, 0, stream>>>(E1B, X2, ei, E, BUFA);
  stats(BUFA, 336, E, 332);
  fold(e2bn0g, e2bn0b, e2W1, e2B1, 332, 152, 336, 160);
  gemm(BUFA, 336, E, BUFB, 336, 160, 1, 1);
  fold(e2bn1g, e2bn1b, e2W2, e2B2, 152, 152, 160, 160);
  gemm(BUFB, 160, E, BUFA, 160, 160, 1, 1);
  fold(e2bn2g, e2bn2b, e2W3, e2B3, 152, 152, 160, 160);
  gemm(BUFA, 160, E, BUFB, 160, 160, 1, 1);
  // SUM/SQ now hold stats of the pre-BN features -> feed ehead.W1 fold.

  // ======== stage F: ehead (lin, lin, relu, lin, relu, lin, lin->1) =======
  fold(e2bn3g, e2bn3b, ehW1, ehB1, 152, 304, 160, 320);
  gemm(BUFB, 160, E, BUFA, 160, 320, 0, 0);
  fold(nullptr, nullptr, ehW2, ehB2, 304, 304, 320, 320);
  gemm(BUFA, 320, E, BUFB, 320, 320, 1, 0);   // relu follows lin2
  fold(nullptr, nullptr, ehW3, ehB3, 304, 304, 320, 320);
  gemm(BUFB, 320, E, BUFA, 320, 320, 1, 0);   // relu follows lin3
  fold(nullptr, nullptr, ehW4, ehB4, 304, 304, 320, 320);
  gemm(BUFA, 320, E, BUFB, 320, 320, 0, 0);
  dot_sigmoid_kernel<<<E, 256, 0, stream>>>(BUFB, 320, 304, E, ehW5, ehB5, outEdge);

  // ======== stage E: nhead (lin, relu, lin, relu, lin, lin->1) ============
  fold(nullptr, nullptr, nhW1, nhB1, 128, 256, 128, 256);
  gemm(X2, 128, N, BUFA, 128, 256, 1, 0);
  fold(nullptr, nullptr, nhW2, nhB2, 256, 256, 256, 256);
  gemm(BUFA, 256, N, BUFB, 256, 256, 1, 0);
  fold(nullptr, nullptr, nhW3, nhB3, 256, 256, 256, 256);
  gemm(BUFB, 256, N, BUFA, 256, 256, 0, 0);
  dot_sigmoid_kernel<<<N, 256, 0, stream>>>(BUFA, 256, 256, N, nhW4, nhB4, outNode);
}